// StateEncoder_49100066128400
// MI455X (gfx1250) — compile-verified
//
#include <hip/hip_runtime.h>
#include <math.h>

// ---------------------------------------------------------------------------
// Reference reduces to:
//   outs[0..2] : 3 x (32,60,1) of exactly 1.0f (softmax over PRED_BITS=1 class)
//   changes    : (32,120,1) = tanh([sem, action] @ gw^T + gb), with sem derived
//                from fixed one-hot rows through sw/sb and the reference's
//                axis-scrambling reshape (p,Bn*T,SEM) -> (Bn,p*T,SEM).
// We compute `changes` with V_WMMA_F32_16X16X4_F32 (exact fp32 matrix path):
// rows tiled by 16, K=8 as two chained K=4 WMMAs, bias preloaded into C.
// ---------------------------------------------------------------------------

typedef __attribute__((ext_vector_type(2))) float v2f;
typedef __attribute__((ext_vector_type(8))) float v8f;

#define BN 32  // batch

__global__ void fill_ones_kernel(float* __restrict__ out, int n4) {
    int i = blockIdx.x * blockDim.x + threadIdx.x;
    if (i < n4) {
        float4 v = make_float4(1.0f, 1.0f, 1.0f, 1.0f);
        reinterpret_cast<float4*>(out)[i] = v;
    }
}

// One wave32 per 16-row tile. Tiles 0..19  -> a=1 branch (T=5,  320 rows).
//                              Tiles 20..119 -> a=2 branch (T=25, 1600 rows).
__global__ void changes_wmma_kernel(const float* __restrict__ action,
                                    const float* __restrict__ sem_w1,
                                    const float* __restrict__ sem_b1,
                                    const float* __restrict__ chg_w1,
                                    const float* __restrict__ chg_b1,
                                    const float* __restrict__ sem_w2,
                                    const float* __restrict__ sem_b2,
                                    const float* __restrict__ chg_w2,
                                    const float* __restrict__ chg_b2,
                                    float* __restrict__ out_changes) {
    const int w    = blockIdx.x;      // 0..119
    const int lane = threadIdx.x;     // 0..31 (wave32)
    const int m    = lane & 15;       // row within tile (A layout), N col (B/C/D)
    const int half = lane >> 4;       // K-half selector for A/B operand regs

    int T, tile, base0, base1, isA2;
    const float *sw, *sb, *gw, *gb;
    if (w < 20) { T = 5;  tile = w;      base0 = 0;  base1 = 10; isA2 = 0;
                  sw = sem_w1; sb = sem_b1; gw = chg_w1; gb = chg_b1; }
    else        { T = 25; tile = w - 20; base0 = 20; base1 = 70; isA2 = 1;
                  sw = sem_w2; sb = sem_b2; gw = chg_w2; gb = chg_b2; }

    // Flat row index L = b*2T + j -> (q, t, b) per the reference's reshape.
    const int r  = tile * 16 + m;
    const int BT = BN * T;
    const int q  = r / BT;
    const int t  = (r % BT) % T;
    const int b  = r / (2 * T);

    // sem[q,t,s] for this lane's two K slots: s = 2*half + {0,1}
    float semv[2];
#pragma unroll
    for (int j = 0; j < 2; ++j) {
        const int s = 2 * half + j;
        float v;
        if (!isA2) {
            // oh = onehot(t) in R^5  ->  sw column t
            v = sw[q * 20 + s * 5 + t] + sb[q * 4 + s];
        } else {
            // oh has ones at t0=t/5, t1=t%5 (single one if equal), in R^10
            const int t0 = t / 5, t1 = t % 5;
            v = sw[q * 40 + s * 10 + t0] + sb[q * 4 + s];
            if (t1 != t0) v += sw[q * 40 + s * 10 + t1];
        }
        semv[j] = tanhf(v);
    }

    // A operand (16x4 f32): lane holds row M=m; regs = K = 2*half + {0,1}
    v2f a_lo, a_hi;
    a_lo.x = semv[0];
    a_lo.y = semv[1];
    a_hi.x = action[b * 4 + 2 * half + 0];   // full[:,4..7] = action_latent[b]
    a_hi.y = action[b * 4 + 2 * half + 1];

    // B operand (4x16 f32): lane holds col N=m; regs = K = 2*half + {0,1}.
    // Only N<2 columns are real (gw is [2 x 8]); rest are zero.
    const int n = m;
    const bool live = (n < 2);
    v2f b_lo, b_hi;
    b_lo.x = live ? gw[n * 8 + 2 * half + 0]     : 0.0f;
    b_lo.y = live ? gw[n * 8 + 2 * half + 1]     : 0.0f;
    b_hi.x = live ? gw[n * 8 + 4 + 2 * half + 0] : 0.0f;
    b_hi.y = live ? gw[n * 8 + 4 + 2 * half + 1] : 0.0f;

    // C preloaded with bias: C[M,n] = gb[n] (same for all 8 row-registers).
    const float cval = live ? gb[n] : 0.0f;
    v8f c;
#pragma unroll
    for (int i = 0; i < 8; ++i) c[i] = cval;

    // D = A_lo*B_lo + A_hi*B_hi + C   (K=8 via two chained K=4 WMMAs)
    c = __builtin_amdgcn_wmma_f32_16x16x4_f32(false, a_lo, false, b_lo,
                                              (short)0, c, false, false);
    c = __builtin_amdgcn_wmma_f32_16x16x4_f32(false, a_hi, false, b_hi,
                                              (short)0, c, false, false);

    // D layout: reg i -> M = i + 8*half, N = lane&15. Scatter tanh(D) to
    // changes[b,120] with slot bases [0,10,20,70] for (a,q') blocks.
    if (live) {
        const int base = (n == 0) ? base0 : base1;
#pragma unroll
        for (int i = 0; i < 8; ++i) {
            const int ro = tile * 16 + i + 8 * half;  // global row
            const int bo = ro / (2 * T);
            const int jo = ro % (2 * T);
            out_changes[bo * 120 + base + jo] = tanhf(c[i]);
        }
    }
}

extern "C" void kernel_launch(void* const* d_in, const int* in_sizes, int n_in,
                              void* d_out, int out_size, void* d_ws, size_t ws_size,
                              hipStream_t stream) {
    (void)in_sizes; (void)n_in; (void)out_size; (void)d_ws; (void)ws_size;

    float* out = (float*)d_out;

    // outs[0..2]: 3 * 32 * 60 = 5760 floats of 1.0f  (1440 float4 stores)
    fill_ones_kernel<<<6, 256, 0, stream>>>(out, 1440);

    // setup_inputs() order: [4]=action_latent, a=1 params at 13..16 (sem_w1,
    // sem_b1, chg_w1, chg_b1), a=2 params at 23..26.
    const float* action = (const float*)d_in[4];
    changes_wmma_kernel<<<120, 32, 0, stream>>>(
        action,
        (const float*)d_in[13], (const float*)d_in[14],
        (const float*)d_in[15], (const float*)d_in[16],
        (const float*)d_in[23], (const float*)d_in[24],
        (const float*)d_in[25], (const float*)d_in[26],
        out + 5760);
}